// EnclaveLayer_90228672955157
// MI455X (gfx1250) — compile-verified
//
#include <hip/hip_runtime.h>

typedef __attribute__((ext_vector_type(2))) float v2f;
typedef __attribute__((ext_vector_type(4))) float v4f;
typedef __attribute__((ext_vector_type(8))) float v8f;

#define DD 64
#define LSTRIDE 68   // padded LDS row stride for logits (floats): banks (4m+c)%64 -> conflict-free
#define NWAVES 8
#define ROWS_PER_BLOCK 128   // 8 waves * 16 rows

// W staged in WMMA B-fragment order:
//   fragment (kc, nt), lane l = 16*half + ln holds
//     W[4*kc + 2*half + 0][16*nt + ln]  and  W[4*kc + 2*half + 1][16*nt + ln]
//   stored contiguously as float2 at sWf[((kc*4 + nt)*32 + l)*2]
__global__ __launch_bounds__(256) void onehot_argmax_gemm(
    const float* __restrict__ x, const float* __restrict__ W,
    const float* __restrict__ bias, float* __restrict__ out)
{
    __shared__ float sWf[DD * DD];                   // 16384 B, fragment-ordered
    __shared__ float sB[DD];                         //   256 B
    __shared__ float sLog[NWAVES][16 * LSTRIDE];     // 34816 B
    __shared__ int   sAmax[NWAVES][16];              //   512 B

    const int tid = threadIdx.x;

    // ---- Stage W into fragment order (coalesced global read, LDS scatter) ----
    #pragma unroll
    for (int s = 0; s < 16; ++s) {
        int idx = tid + 256 * s;        // flat element of W, 4096 total
        int k = idx >> 6;               // W row
        int c = idx & 63;               // W col
        int kc   = k >> 2;
        int rem  = k & 3;
        int half = rem >> 1;
        int p    = rem & 1;
        int nt   = c >> 4;
        int ln   = c & 15;
        int lanei = 16 * half + ln;
        sWf[((kc * 4 + nt) * 32 + lanei) * 2 + p] = W[idx];
    }
    if (tid < DD) sB[tid] = bias[tid];
    __syncthreads();

    const int wave = tid >> 5;
    const int lane = tid & 31;
    const int half = lane >> 4;   // 0: K=0,1 ; 1: K=2,3 (A layout)
    const int ln   = lane & 15;

    const long long rowBase = (long long)blockIdx.x * ROWS_PER_BLOCK + (long long)wave * 16;
    const float* __restrict__ xt = x   + rowBase * DD;
    float*       __restrict__ ot = out + rowBase * DD;

    // ---- Load A fragments: 16 K-chunks of 16x4 fp32 ----
    // lane(half,ln): VGPR0 = x[ln][4*kc + 2*half], VGPR1 = x[ln][4*kc + 2*half + 1]
    v2f a[16];
    {
        const float* xr = xt + ln * DD + 2 * half;
        #pragma unroll
        for (int kc = 0; kc < 16; ++kc)
            a[kc] = *(const v2f*)(xr + 4 * kc);
    }

    float* myLog = sLog[wave];

    // ---- 4 N-tiles of 16 columns; 16 WMMA_F32_16X16X4_F32 each ----
    #pragma unroll
    for (int nt = 0; nt < 4; ++nt) {
        const float bv = sB[nt * 16 + ln];
        v8f acc = { bv, bv, bv, bv, bv, bv, bv, bv };   // C[m][n] starts at b[n]

        #pragma unroll
        for (int kc = 0; kc < 16; ++kc) {
            v2f bb = *(const v2f*)(&sWf[((kc * 4 + nt) * 32 + lane) * 2]);
            acc = __builtin_amdgcn_wmma_f32_16x16x4_f32(
                false, a[kc], false, bb, (short)0, acc, false, false);
        }

        // C/D layout: VGPR i -> row (i + 8*half), col (nt*16 + ln)
        #pragma unroll
        for (int i = 0; i < 8; ++i)
            myLog[(i + 8 * half) * LSTRIDE + nt * 16 + ln] = acc[i];
    }
    __syncthreads();

    // ---- Argmax per row (first max wins, ascending scan with strict >) ----
    if (lane < 16) {
        const float* lr = myLog + lane * LSTRIDE;
        float best = lr[0];
        int   bi   = 0;
        #pragma unroll 8
        for (int c = 1; c < DD; ++c) {
            float v = lr[c];
            if (v > best) { best = v; bi = c; }
        }
        sAmax[wave][lane] = bi;
    }
    __syncthreads();

    // ---- Write one-hot tile (16x64), coalesced float4 stores ----
    #pragma unroll
    for (int j = 0; j < 8; ++j) {
        int q  = lane + 32 * j;     // float4 index in tile (256 total)
        int r  = q >> 4;
        int c4 = q & 15;
        int hot = sAmax[wave][r];
        int c0  = c4 * 4;
        v4f o;
        o.x = (c0 + 0 == hot) ? 1.0f : 0.0f;
        o.y = (c0 + 1 == hot) ? 1.0f : 0.0f;
        o.z = (c0 + 2 == hot) ? 1.0f : 0.0f;
        o.w = (c0 + 3 == hot) ? 1.0f : 0.0f;
        *(v4f*)(ot + r * DD + c0) = o;
    }
}

extern "C" void kernel_launch(void* const* d_in, const int* in_sizes, int n_in,
                              void* d_out, int out_size, void* d_ws, size_t ws_size,
                              hipStream_t stream) {
    const float* x = (const float*)d_in[0];   // [B, 64]
    const float* W = (const float*)d_in[1];   // [64, 64]
    const float* b = (const float*)d_in[2];   // [64]
    float* out = (float*)d_out;               // [B, 64]

    const long long Brows = (long long)in_sizes[0] / DD;   // 2097152
    const int grid = (int)((Brows + ROWS_PER_BLOCK - 1) / ROWS_PER_BLOCK);

    onehot_argmax_gemm<<<grid, 256, 0, stream>>>(x, W, b, out);
}